// Model_14130442404380
// MI455X (gfx1250) — compile-verified
//
#include <hip/hip_runtime.h>
#include <math.h>

typedef __attribute__((ext_vector_type(2))) float v2f;
typedef __attribute__((ext_vector_type(8))) float v8f;

#define ACT_NONE 0
#define ACT_SILU 1
#define ACT_SOFTPLUS 2

struct MambaP { const float *A_log,*Dp,*conv_b,*conv_w,*dt_b,*dt_w,*in_w,*out_w,*xproj_w; };

__device__ __forceinline__ float silu_f(float x){ return x * (1.f/(1.f+__expf(-x))); }
__device__ __forceinline__ float act_f(float v, int act){
    if (act == ACT_SILU)          return silu_f(v);
    if (act == ACT_SOFTPLUS)      return (v > 20.f) ? v : log1pf(__expf(v));
    return v;
}

// ---------------------------------------------------------------------------
// Tiled GEMM: C[M,N] = act(A[M,K](lda) @ W[N,K]^T + bias) + beta*C
// Requires M%64==0 && N%32==0 && K%4==0, and even lda/K (true for all GEMMs
// in this model) so every WMMA fragment is an aligned 8-byte pair.
// One wave owns a 64x32 macro-tile (4x2 grid of 16x16 WMMA tiles):
//   - 64 accumulator VGPRs, no spills
//   - uniform SGPR base + 32-bit lane offsets -> global_load saddr form
//   - per K-step: 6 global_load_b64 feed 8 v_wmma_f32_16x16x4_f32
//   - no divergence before stores -> EXEC all-ones at every WMMA
// ---------------------------------------------------------------------------
__global__ __launch_bounds__(128)
void gemm_wmma_tiled_kernel(const float* __restrict__ A, int lda,
                            const float* __restrict__ W,
                            const float* __restrict__ bias,
                            float* __restrict__ C,
                            int M, int N, int K, int act, float beta)
{
    const int lane = threadIdx.x & 31;
    const int tn   = blockIdx.x * blockDim.y + threadIdx.y;   // 32-wide N tile
    const int tmi  = blockIdx.y;                              // 64-wide M tile
    const int n0 = tn * 32, m0 = tmi * 64;
    if (m0 >= M || n0 >= N) return;          // wave-uniform exit

    const int half = lane >> 4;              // K sub-offset 0 / 2
    const int lid  = lane & 15;

    unsigned aoff[4], boff[2];
#pragma unroll
    for (int i = 0; i < 4; ++i)
        aoff[i] = (unsigned)(m0 + 16*i + lid) * (unsigned)lda + 2u*half;
#pragma unroll
    for (int j = 0; j < 2; ++j)
        boff[j] = (unsigned)(n0 + 16*j + lid) * (unsigned)K + 2u*half;

    v8f acc[4][2];
#pragma unroll
    for (int i = 0; i < 4; ++i)
#pragma unroll
        for (int j = 0; j < 2; ++j)
            acc[i][j] = (v8f){0.f,0.f,0.f,0.f,0.f,0.f,0.f,0.f};

    for (int kk = 0; kk < K; kk += 4) {
        v2f a[4], b[2];
#pragma unroll
        for (int i = 0; i < 4; ++i)
            a[i] = *(const v2f*)(A + aoff[i] + kk);           // global_load_b64
#pragma unroll
        for (int j = 0; j < 2; ++j)
            b[j] = *(const v2f*)(W + boff[j] + kk);           // global_load_b64
#pragma unroll
        for (int i = 0; i < 4; ++i)
#pragma unroll
            for (int j = 0; j < 2; ++j)
                acc[i][j] = __builtin_amdgcn_wmma_f32_16x16x4_f32(
                    false, a[i], false, b[j], (short)0, acc[i][j], false, false);
    }

#pragma unroll
    for (int j = 0; j < 2; ++j) {
        int col = n0 + 16*j + lid;
        float bv = bias ? bias[col] : 0.f;
#pragma unroll
        for (int i = 0; i < 4; ++i) {
#pragma unroll
            for (int r = 0; r < 8; ++r) {
                unsigned row = (unsigned)(m0 + 16*i + 8*half + r);
                unsigned co  = row * (unsigned)N + (unsigned)col;
                float v = act_f(acc[i][j][r] + bv, act);
                if (beta != 0.f) v += beta * C[co];
                C[co] = v;
            }
        }
    }
}

// ---------------------------------------------------------------------------
// Fallback GEMM for ragged N (36): one wave per 16x16 tile.
// Ragged columns handled by clamping the row index to a valid row and
// scaling the loaded value by a 0/1 mask -> loads stay unconditional,
// EXEC stays full through every WMMA (no saveexec in the hot loop).
// ---------------------------------------------------------------------------
__global__ __launch_bounds__(128)
void gemm_wmma_kernel(const float* __restrict__ A, int lda,
                      const float* __restrict__ W,
                      const float* __restrict__ bias,
                      float* __restrict__ C,
                      int M, int N, int K, int act, float beta)
{
    const int lane = threadIdx.x & 31;
    const int tn   = blockIdx.x * blockDim.y + threadIdx.y;
    const int tmi  = blockIdx.y;
    const int n0 = tn * 16, m0 = tmi * 16;
    if (m0 >= M || n0 >= N) return;          // wave-uniform exit

    const int  half   = lane >> 4;
    const int  lid    = lane & 15;
    const int  arow   = m0 + lid;
    const int  bcol   = n0 + lid;
    const bool bvalid = (bcol < N);
    const float bscale = bvalid ? 1.f : 0.f;

    const unsigned aoff = (unsigned)arow * (unsigned)lda + 2u*half;
    const unsigned boff = (unsigned)(bvalid ? bcol : (N-1)) * (unsigned)K + 2u*half;

    v8f acc = {0.f,0.f,0.f,0.f,0.f,0.f,0.f,0.f};
    for (int kk = 0; kk < K; kk += 4) {
        v2f a = *(const v2f*)(A + aoff + kk);                 // global_load_b64
        v2f b = *(const v2f*)(W + boff + kk);                 // global_load_b64
        b.x *= bscale;                                        // unconditional load + mul
        b.y *= bscale;
        acc = __builtin_amdgcn_wmma_f32_16x16x4_f32(
                  false, a, false, b, (short)0, acc, false, false);
    }

    if (!bvalid) return;                     // divergence only after last WMMA
    float bv = bias ? bias[bcol] : 0.f;
#pragma unroll
    for (int i = 0; i < 8; ++i) {
        int row = m0 + i + 8*half;
        if (row >= M) continue;
        float v = act_f(acc[i] + bv, act);
        unsigned co = (unsigned)row * (unsigned)N + (unsigned)bcol;
        if (beta != 0.f) v += beta * C[co];
        C[co] = v;
    }
}

// ---------------------------------------------------------------------------
// Depthwise causal conv (k taps) + bias + SiLU.  xz layout (B*L, 2*di), xi
// part is the first di columns.  Out: xi_act (B*L, di).
// ---------------------------------------------------------------------------
__global__ void conv_silu_kernel(const float* __restrict__ xz, float* __restrict__ xi,
                                 const float* __restrict__ cw, const float* __restrict__ cb,
                                 int Bn, int L, int di, int k)
{
    long idx = (long)blockIdx.x * blockDim.x + threadIdx.x;
    long tot = (long)Bn * L * di;
    if (idx >= tot) return;
    int  d  = idx % di;
    long bl = idx / di;
    int  l  = bl % L;
    int  b  = bl / L;
    long base = (long)b * L * (2L*di);
    float s = cb[d];
    for (int j = 0; j < k; ++j) {
        int ls = l - (k-1) + j;
        if (ls >= 0) s += xz[base + (long)ls*2*di + d] * cw[d*k + j];
    }
    xi[idx] = silu_f(s);
}

// ---------------------------------------------------------------------------
// Selective scan: one thread per (b,d); h[16] in registers; reverse flag
// implements the flipped mamba (exact for k==1 convs).  Fuses +xi*D and
// *silu(z) epilogue.  Bm/Cm read from dbc rows at offset dtr / dtr+16.
// ---------------------------------------------------------------------------
__global__ void scan_kernel(const float* __restrict__ dt, const float* __restrict__ xi,
                            const float* __restrict__ dbc, const float* __restrict__ A_log,
                            const float* __restrict__ Dp, const float* __restrict__ xz,
                            float* __restrict__ out,
                            int Bn, int L, int di, int dtr, int reverse)
{
    int idx = blockIdx.x * blockDim.x + threadIdx.x;
    if (idx >= Bn * di) return;
    int b = idx / di, d = idx % di;
    int stride = dtr + 32;

    float A[16];
#pragma unroll
    for (int n = 0; n < 16; ++n) A[n] = -__expf(A_log[d*16 + n]);
    float Dd = Dp[d];
    float h[16];
#pragma unroll
    for (int n = 0; n < 16; ++n) h[n] = 0.f;

    for (int step = 0; step < L; ++step) {
        int  l   = reverse ? (L - 1 - step) : step;
        long row = (long)b * L + l;
        float dtv = dt[row*di + d];
        float xv  = xi[row*di + d];
        const float* bc = dbc + row*stride + dtr;   // Bm[0..15], Cm[16..31]
        float y = 0.f;
#pragma unroll
        for (int n = 0; n < 16; ++n) {
            float dA = __expf(dtv * A[n]);
            h[n] = dA * h[n] + dtv * bc[n] * xv;
            y   += h[n] * bc[16 + n];
        }
        y += xv * Dd;
        float zv = xz[row*(2L*di) + di + d];
        y *= silu_f(zv);
        out[row*di + d] = y;
    }
}

// ---------------------------------------------------------------------------
__global__ void layernorm_kernel(const float* __restrict__ x, const float* __restrict__ w,
                                 const float* __restrict__ b, float* __restrict__ y, int cols)
{
    int row = blockIdx.x;
    const float* xr = x + (long)row * cols;
    __shared__ float red[256];
    float s = 0.f;
    for (int c = threadIdx.x; c < cols; c += 256) s += xr[c];
    red[threadIdx.x] = s; __syncthreads();
    for (int o = 128; o; o >>= 1) { if (threadIdx.x < o) red[threadIdx.x] += red[threadIdx.x+o]; __syncthreads(); }
    float mean = red[0] / cols; __syncthreads();
    float v = 0.f;
    for (int c = threadIdx.x; c < cols; c += 256) { float dl = xr[c]-mean; v += dl*dl; }
    red[threadIdx.x] = v; __syncthreads();
    for (int o = 128; o; o >>= 1) { if (threadIdx.x < o) red[threadIdx.x] += red[threadIdx.x+o]; __syncthreads(); }
    float rstd = rsqrtf(red[0]/cols + 1e-5f);
    float* yr = y + (long)row * cols;
    for (int c = threadIdx.x; c < cols; c += 256) yr[c] = (xr[c]-mean)*rstd*w[c] + b[c];
}

__global__ void meanstd_kernel(const float* __restrict__ x, float* __restrict__ means,
                               float* __restrict__ stdev, int T, int C)
{
    int bc = blockIdx.x;
    int b = bc / C, c = bc % C;
    const float* xp = x + (long)b*T*C + c;
    __shared__ float red[256];
    float s = 0.f;
    for (int t = threadIdx.x; t < T; t += 256) s += xp[(long)t*C];
    red[threadIdx.x] = s; __syncthreads();
    for (int o = 128; o; o >>= 1) { if (threadIdx.x < o) red[threadIdx.x] += red[threadIdx.x+o]; __syncthreads(); }
    float mean = red[0] / T; __syncthreads();
    float v = 0.f;
    for (int t = threadIdx.x; t < T; t += 256) { float d = xp[(long)t*C]-mean; v += d*d; }
    red[threadIdx.x] = v; __syncthreads();
    for (int o = 128; o; o >>= 1) { if (threadIdx.x < o) red[threadIdx.x] += red[threadIdx.x+o]; __syncthreads(); }
    if (threadIdx.x == 0) { means[bc] = mean; stdev[bc] = sqrtf(red[0]/T + 1e-5f); }
}

__global__ void norm_transpose_kernel(const float* __restrict__ x, const float* __restrict__ means,
                                      const float* __restrict__ stdev, float* __restrict__ xt,
                                      int Bn, int T, int C)
{
    long idx = (long)blockIdx.x * blockDim.x + threadIdx.x;
    long tot = (long)Bn * C * T;
    if (idx >= tot) return;
    int  t  = idx % T;
    long bc = idx / T;
    int  c = bc % C, b = bc / C;
    xt[idx] = (x[((long)b*T + t)*C + c] - means[bc]) / stdev[bc];
}

__global__ void decomp_kernel(const float* __restrict__ x, float* __restrict__ seas,
                              float* __restrict__ trend, long rows, int cols, int kdec)
{
    long idx = (long)blockIdx.x * blockDim.x + threadIdx.x;
    long tot = rows * (long)cols;
    if (idx >= tot) return;
    int  col = idx % cols;
    long row = idx / cols;
    const float* xr = x + row * cols;
    int pad = (kdec - 1) / 2;
    float s = 0.f;
    for (int j = -pad; j <= pad; ++j) {
        int jj = col + j;
        jj = jj < 0 ? 0 : (jj >= cols ? cols - 1 : jj);   // edge-replicate == clamp
        s += xr[jj];
    }
    float mm = s / kdec;
    trend[idx] = mm;
    seas[idx]  = x[idx] - mm;
}

__global__ void transpose_kernel(const float* __restrict__ in, float* __restrict__ out,
                                 int Bn, int C, int Dm)   // (B,C,Dm) -> (B,Dm,C)
{
    long idx = (long)blockIdx.x * blockDim.x + threadIdx.x;
    long tot = (long)Bn * C * Dm;
    if (idx >= tot) return;
    int  c  = idx % C;
    long bj = idx / C;
    int  j = bj % Dm, b = bj / Dm;
    out[idx] = in[((long)b*C + c)*Dm + j];
}

__global__ void combine_kernel(const float* __restrict__ y1t, const float* __restrict__ y0,
                               const float* __restrict__ xres, float* __restrict__ out,
                               int Bn, int C, int Dm)   // out[b,c,j] = y1t[b,j,c]*y0[b,c,j]+xres[b,c,j]
{
    long idx = (long)blockIdx.x * blockDim.x + threadIdx.x;
    long tot = (long)Bn * C * Dm;
    if (idx >= tot) return;
    int  j  = idx % Dm;
    long bc = idx / Dm;
    int  c = bc % C, b = bc / C;
    out[idx] = y1t[((long)b*Dm + j)*C + c] * y0[idx] + xres[idx];
}

__global__ void add_kernel(const float* __restrict__ a, const float* __restrict__ b,
                           float* __restrict__ o, long n)
{
    long idx = (long)blockIdx.x * blockDim.x + threadIdx.x;
    if (idx < n) o[idx] = a[idx] + b[idx];
}

__global__ void final_kernel(const float* __restrict__ dec, const float* __restrict__ stdev,
                             const float* __restrict__ means, float* __restrict__ out,
                             int Bn, int P, int C)   // out[b,t,c] = dec[(b*C+c)*P+t]*std+mean
{
    long idx = (long)blockIdx.x * blockDim.x + threadIdx.x;
    long tot = (long)Bn * P * C;
    if (idx >= tot) return;
    int  c  = idx % C;
    long bt = idx / C;
    int  t = bt % P, b = bt / P;
    int bc = b*C + c;
    out[idx] = dec[((long)bc)*P + t] * stdev[bc] + means[bc];
}

// ===========================================================================
extern "C" void kernel_launch(void* const* d_in, const int* in_sizes, int n_in,
                              void* d_out, int out_size, void* d_ws, size_t ws_size,
                              hipStream_t stream)
{
    (void)in_sizes; (void)out_size; (void)ws_size;
    const int B = 16, T = 512, CIN = 64, DM = 512, PRED = 96;

    // ---- parameter leaves in JAX pytree (sorted-dict-key) order ----------
    const int NLEAF = 138;
    long sizes[NLEAF];
    int li = 0;
    auto push_mamba_sz = [&](int D, int di, int dtr, int k) {
        sizes[li++] = (long)di*16;        // A_log
        sizes[li++] = di;                 // D
        sizes[li++] = di;                 // conv_b
        sizes[li++] = (long)di*k;         // conv_w
        sizes[li++] = di;                 // dt_b
        sizes[li++] = (long)di*dtr;       // dt_w
        sizes[li++] = 2L*di*D;            // in_w
        sizes[li++] = (long)D*di;         // out_w
        sizes[li++] = (long)(dtr+32)*di;  // xproj_w
    };
    auto push_enc_sz = [&]() {
        push_mamba_sz(512,1024,32,1);     // cm_b
        push_mamba_sz(512,1024,32,1);     // cm_f
        sizes[li++] = 512;                // n1b
        sizes[li++] = 512;                // n1w
        push_mamba_sz(64,128,4,4);        // tm
    };
    sizes[li++] = 512;                    // emb_b
    sizes[li++] = 262144;                 // emb_w
    push_mamba_sz(512,1024,32,1);         // pre_b
    push_mamba_sz(512,1024,32,1);         // pre_f
    sizes[li++] = 96;                     // proj_b
    sizes[li++] = 49152;                  // proj_w
    push_enc_sz(); push_enc_sz();         // seasonal[0], seasonal[1]
    push_enc_sz(); push_enc_sz();         // trend[0], trend[1]

    const float* P[NLEAF];
    if (n_in >= 4 + NLEAF) {
        for (int j = 0; j < NLEAF; ++j) P[j] = (const float*)d_in[4+j];
    } else {
        const float* base = (const float*)d_in[4];
        long off = 0;
        for (int j = 0; j < NLEAF; ++j) { P[j] = base + off; off += sizes[j]; }
    }
    auto mk = [&](int b)->MambaP {
        return { P[b],P[b+1],P[b+2],P[b+3],P[b+4],P[b+5],P[b+6],P[b+7],P[b+8] };
    };

    // ---- workspace layout -------------------------------------------------
    float* ws = (float*)d_ws;
    long o = 0;
    float* means = ws+o; o += 1024;
    float* stdev = ws+o; o += 1024;
    float* xt    = ws+o; o += 524288;     // normalized transposed input
    float* bufA  = ws+o; o += 524288;     // embed -> seasonal stream
    float* bufS  = ws+o; o += 524288;     // pre-bimamba out / seasonal+trend sum
    float* bufT  = ws+o; o += 524288;     // trend stream
    float* y0    = ws+o; o += 524288;
    float* y0t   = ws+o; o += 524288;
    float* y1t   = ws+o; o += 524288;
    float* xz    = ws+o; o += 2097152;    // (B*L, 2*di)  (both mamba sizes == 2M)
    float* xi    = ws+o; o += 1048576;    // (B*L, di)
    float* dbc   = ws+o; o += 294912;     // (B*L, dtr+32), max for tm
    float* dtb   = ws+o; o += 1048576;    // softplus(dt) (B*L, di)
    float* scano = ws+o; o += 1048576;    // scan output (B*L, di)
    float* dec   = ws+o; o += 98304;

    auto gemm = [&](const float* A, int lda, const float* W, const float* bias,
                    float* C, int M, int N, int K, int act, float beta) {
        if ((M % 64) == 0 && (N % 32) == 0) {
            int Mt = M/64, Nt = N/32;
            dim3 grid((Nt+3)/4, Mt), block(32, 4);
            gemm_wmma_tiled_kernel<<<grid, block, 0, stream>>>(A, lda, W, bias, C, M, N, K, act, beta);
        } else {
            int Mt = (M+15)/16, Nt = (N+15)/16;
            dim3 grid((Nt+3)/4, Mt), block(32, 4);
            gemm_wmma_kernel<<<grid, block, 0, stream>>>(A, lda, W, bias, C, M, N, K, act, beta);
        }
    };

    auto run_mamba = [&](const MambaP& mp, const float* x, float* out,
                         int L, int D, int di, int dtr, int k, int reverse, float beta) {
        int M = B * L;
        gemm(x, D, mp.in_w, nullptr, xz, M, 2*di, D, ACT_NONE, 0.f);
        long tot = (long)M * di;
        conv_silu_kernel<<<dim3((unsigned)((tot+255)/256)), 256, 0, stream>>>(
            xz, xi, mp.conv_w, mp.conv_b, B, L, di, k);
        gemm(xi, di, mp.xproj_w, nullptr, dbc, M, dtr+32, di, ACT_NONE, 0.f);
        gemm(dbc, dtr+32, mp.dt_w, mp.dt_b, dtb, M, di, dtr, ACT_SOFTPLUS, 0.f);
        int st = B * di;
        scan_kernel<<<dim3((st+255)/256), 256, 0, stream>>>(
            dtb, xi, dbc, mp.A_log, mp.Dp, xz, scano, B, L, di, dtr, reverse);
        gemm(scano, di, mp.out_w, nullptr, out, M, D, di, ACT_NONE, beta);
    };

    const size_t xbytes = (size_t)B*CIN*DM*sizeof(float);

    auto run_encoder = [&](int e, float* xbuf) {
        MambaP cmb = mk(e), cmf = mk(e+9), tm = mk(e+20);
        const float* n1b = P[e+18];
        const float* n1w = P[e+19];
        // y0 = x + mamba_f(x) + flip(mamba_b(flip(x)))
        hipMemcpyAsync(y0, xbuf, xbytes, hipMemcpyDeviceToDevice, stream);
        run_mamba(cmf, xbuf, y0, 64, DM, 1024, 32, 1, /*rev*/0, /*beta*/1.f);
        run_mamba(cmb, xbuf, y0, 64, DM, 1024, 32, 1, /*rev*/1, /*beta*/1.f);
        layernorm_kernel<<<B*CIN, 256, 0, stream>>>(y0, n1w, n1b, y0, DM);
        // y1 = mamba_tm(y0^T)^T ;  y2 = y1*y0 + x ; LN
        long tot = (long)B*CIN*DM;
        transpose_kernel<<<dim3((unsigned)((tot+255)/256)), 256, 0, stream>>>(y0, y0t, B, CIN, DM);
        run_mamba(tm, y0t, y1t, 512, 64, 128, 4, 4, /*rev*/0, /*beta*/0.f);
        combine_kernel<<<dim3((unsigned)((tot+255)/256)), 256, 0, stream>>>(y1t, y0, xbuf, xbuf, B, CIN, DM);
        layernorm_kernel<<<B*CIN, 256, 0, stream>>>(xbuf, n1w, n1b, xbuf, DM);
    };

    // ---- forward pass -----------------------------------------------------
    const float* x_enc = (const float*)d_in[0];
    meanstd_kernel<<<B*CIN, 256, 0, stream>>>(x_enc, means, stdev, T, CIN);
    {
        long tot = (long)B*CIN*T;
        norm_transpose_kernel<<<dim3((unsigned)((tot+255)/256)), 256, 0, stream>>>(
            x_enc, means, stdev, xt, B, T, CIN);
    }
    // embedding: (B*64, 512) @ emb_w^T + emb_b   [P[1]=emb_w, P[0]=emb_b]
    gemm(xt, T, P[1], P[0], bufA, B*CIN, DM, T, ACT_NONE, 0.f);
    // pre bimamba (no residual)  [pre_b at leaf 2, pre_f at leaf 11]
    run_mamba(mk(11), bufA, bufS, 64, DM, 1024, 32, 1, 0, 0.f);
    run_mamba(mk(2),  bufA, bufS, 64, DM, 1024, 32, 1, 1, 1.f);
    // series decomposition along d_model
    {
        long tot = (long)B*CIN*DM;
        decomp_kernel<<<dim3((unsigned)((tot+255)/256)), 256, 0, stream>>>(
            bufS, bufA /*seasonal*/, bufT /*trend*/, (long)B*CIN, DM, 13);
    }
    // encoder stacks (leaf bases: seasonal0=22, seasonal1=51, trend0=80, trend1=109)
    run_encoder(22, bufA);
    run_encoder(51, bufA);
    run_encoder(80, bufT);
    run_encoder(109, bufT);
    // x = seasonal + trend
    {
        long tot = (long)B*CIN*DM;
        add_kernel<<<dim3((unsigned)((tot+255)/256)), 256, 0, stream>>>(bufA, bufT, bufS, tot);
    }
    // projection: (B*64,512) @ proj_w^T + proj_b   [P[21]=proj_w, P[20]=proj_b]
    gemm(bufS, DM, P[21], P[20], dec, B*CIN, PRED, DM, ACT_NONE, 0.f);
    // transpose + denormalize -> d_out (16, 96, 64)
    {
        long tot = (long)B*PRED*CIN;
        final_kernel<<<dim3((unsigned)((tot+255)/256)), 256, 0, stream>>>(
            dec, stdev, means, (float*)d_out, B, PRED, CIN);
    }
}